// Render_43173011259673
// MI455X (gfx1250) — compile-verified
//
#include <hip/hip_runtime.h>
#include <math.h>

// ---------------------------------------------------------------------------
// Problem constants (from reference):
//   B=2 batches, V=512 vertices, T=256 triangles, H=W=256, L=2 lights
//   FOV_Y=40, NEAR=0.01, FAR=10
// ---------------------------------------------------------------------------

typedef float v2f __attribute__((ext_vector_type(2)));
typedef float v8f __attribute__((ext_vector_type(8)));
typedef int   v4i __attribute__((vector_size(16)));

#define IMG 256
#define NB  2
#define NV  512
#define NT  256
#define NL  2

#if defined(__AMDGCN__) && \
    __has_builtin(__builtin_amdgcn_global_load_async_to_lds_b128) && \
    __has_builtin(__builtin_amdgcn_s_wait_asynccnt)
#define USE_ASYNC_LDS 1
typedef __attribute__((address_space(1))) v4i* gv4i_p;   // global (prints __device__)
typedef __attribute__((address_space(3))) v4i* lv4i_p;   // LDS    (prints __shared__)
#else
#define USE_ASYNC_LDS 0
#endif

// ---------------------------------------------------------------------------
// Kernel 1: camera setup + vertex transform to clip space.
// Each of B*V threads redundantly builds mvp (tiny) and transforms one vertex.
// clip layout: [b*V + v][4]
// ---------------------------------------------------------------------------
__global__ void k_transform(const float* __restrict__ vertices,
                            const float* __restrict__ eye,
                            const float* __restrict__ center,
                            const float* __restrict__ wup,
                            float* __restrict__ clip)
{
    int idx = blockIdx.x * blockDim.x + threadIdx.x;
    if (idx >= NB * NV) return;

    float ex = eye[0], ey = eye[1], ez = eye[2];
    float fx = center[0] - ex, fy = center[1] - ey, fz = center[2] - ez;
    float fl = sqrtf(fx*fx + fy*fy + fz*fz);
    fx /= fl; fy /= fl; fz /= fl;
    float ux = wup[0], uy = wup[1], uz = wup[2];
    // side = cross(fwd, up)
    float sx = fy*uz - fz*uy;
    float sy = fz*ux - fx*uz;
    float sz = fx*uy - fy*ux;
    float sl = sqrtf(sx*sx + sy*sy + sz*sz);
    sx /= sl; sy /= sl; sz /= sl;
    // up2 = cross(side, fwd)
    float u2x = sy*fz - sz*fy;
    float u2y = sz*fx - sx*fz;
    float u2z = sx*fy - sy*fx;
    // t = -R @ eye   (R rows: side, up2, -fwd)
    float tx = -(sx*ex + sy*ey + sz*ez);
    float ty = -(u2x*ex + u2y*ey + u2z*ez);
    float tz =  (fx*ex + fy*ey + fz*ez);

    const float focal = 1.0f / tanf((40.0f * 0.5f) * (3.14159265358979323846f / 180.0f));
    const float nearf = 0.01f, farf = 10.0f;
    const float p22 = (nearf + farf) / (nearf - farf);
    const float p23 = 2.0f * nearf * farf / (nearf - farf);

    float v0 = vertices[idx*3 + 0];
    float v1 = vertices[idx*3 + 1];
    float v2 = vertices[idx*3 + 2];

    // view rows: r0=(s,tx), r1=(u2,ty), r2=(-f,tz), r3=(0,0,0,1); aspect == 1
    float r0 =  sx*v0 +  sy*v1 +  sz*v2 + tx;
    float r1 = u2x*v0 + u2y*v1 + u2z*v2 + ty;
    float r2 = -fx*v0 -  fy*v1 -  fz*v2 + tz;

    clip[idx*4 + 0] = focal * r0;
    clip[idx*4 + 1] = focal * r1;
    clip[idx*4 + 2] = p22 * r2 + p23;
    clip[idx*4 + 3] = -r2;
}

// ---------------------------------------------------------------------------
// Kernel 2: per-triangle setup. Builds the 4x16-per-group WMMA "B" matrix
// columns for each triangle: cols (b0,b1,b2,z). Storage per triangle is 16
// floats in [col i][row k] order so the raster kernel can ds_load_b64 the two
// rows it needs (k, k+1 adjacent).
//   wsB[(b*T + t)*16 + i*4 + k] = (i<3) ? inv[i][k] : zcol[k]   (k<3; k==3 -> 0)
//   wsW[(b*T + t)*4 + k] = w_clip[k]
// Invalid triangles: b0 column forced to (0,0,-1,0) => bary0 == -1 always.
// ---------------------------------------------------------------------------
__global__ void k_trisetup(const float* __restrict__ clip,
                           const int*   __restrict__ faces,
                           float* __restrict__ wsB,
                           float* __restrict__ wsW)
{
    int idx = blockIdx.x * blockDim.x + threadIdx.x;
    if (idx >= NB * NT) return;
    int b = idx / NT;
    int t = idx - b * NT;

    int i0 = faces[t*3 + 0];
    int i1 = faces[t*3 + 1];
    int i2 = faces[t*3 + 2];

    const float* c0 = clip + (b*NV + i0)*4;
    const float* c1 = clip + (b*NV + i1)*4;
    const float* c2 = clip + (b*NV + i2)*4;

    float w0 = c0[3], w1 = c1[3], w2 = c2[3];
    float x0 = c0[0]/w0, y0 = c0[1]/w0, z0 = c0[2]/w0;
    float x1 = c1[0]/w1, y1 = c1[1]/w1, z1 = c1[2]/w1;
    float x2 = c2[0]/w2, y2 = c2[1]/w2, z2 = c2[2]/w2;

    // m = [[x0,x1,x2],[y0,y1,y2],[1,1,1]]
    float ma = x0, mb = x1, mc = x2;
    float md = y0, me = y1, mf = y2;
    const float mg = 1.0f, mh = 1.0f, mi = 1.0f;
    float A  = me*mi - mf*mh;
    float Bv = mf*mg - md*mi;
    float C  = md*mh - me*mg;
    float D  = mc*mh - mb*mi;
    float E  = ma*mi - mc*mg;
    float F  = mb*mg - ma*mh;
    float G  = mb*mf - mc*me;
    float Hc = mc*md - ma*mf;
    float I  = ma*me - mb*md;
    float det  = ma*A + mb*Bv + mc*C;
    float safe = (fabsf(det) > 1e-9f) ? det : 1.0f;

    float inv[3][3];
    inv[0][0] = A  / safe; inv[0][1] = D  / safe; inv[0][2] = G  / safe;
    inv[1][0] = Bv / safe; inv[1][1] = E  / safe; inv[1][2] = Hc / safe;
    inv[2][0] = C  / safe; inv[2][1] = F  / safe; inv[2][2] = I  / safe;

    bool valid = (fabsf(det) > 1e-9f) && (w0 > 1e-6f) && (w1 > 1e-6f) && (w2 > 1e-6f);

    // zcol[j] = sum_i z_i * inv[i][j]   (so z_pix = p . zcol)
    float zc0 = z0*inv[0][0] + z1*inv[1][0] + z2*inv[2][0];
    float zc1 = z0*inv[0][1] + z1*inv[1][1] + z2*inv[2][1];
    float zc2 = z0*inv[0][2] + z1*inv[1][2] + z2*inv[2][2];

    float Bt[16];
    #pragma unroll
    for (int i = 0; i < 3; ++i) {
        Bt[i*4 + 0] = inv[i][0];
        Bt[i*4 + 1] = inv[i][1];
        Bt[i*4 + 2] = inv[i][2];
        Bt[i*4 + 3] = 0.0f;
    }
    Bt[12] = zc0; Bt[13] = zc1; Bt[14] = zc2; Bt[15] = 0.0f;

    if (!valid) { Bt[0] = 0.0f; Bt[1] = 0.0f; Bt[2] = -1.0f; Bt[3] = 0.0f; }

    float* o = wsB + idx*16;
    #pragma unroll
    for (int j = 0; j < 16; ++j) o[j] = Bt[j];

    float* ow = wsW + idx*4;
    ow[0] = w0; ow[1] = w1; ow[2] = w2; ow[3] = 0.0f;
}

// ---------------------------------------------------------------------------
// Kernel 3: fused raster (WMMA z-buffer) + perspective-correct shading.
// Block = 256 threads = 8 waves; each wave owns 16 consecutive pixels.
// Grid = NB * (IMG*IMG/128) blocks. Triangle matrices staged in LDS
// (async global->LDS path when the toolchain exposes it).
//
// Per 4-triangle group, one V_WMMA_F32_16X16X4_F32 computes a 16x16 tile:
//   rows   M = 16 pixels, A row = (px, py, 1, 0)
//   cols   N = 4*t_local + {b0, b1, b2, z}
// Coverage via ballot over (D >= 0) on the 3 bary lanes of each group,
// z via shfl from lane base+3; running (zmin, argmin-first) per lane slot,
// final shfl_xor(4,8) reduction across the 4 triangle slots with
// lowest-index tie-break to match jnp.argmin.
// ---------------------------------------------------------------------------
__global__ void __launch_bounds__(256)
k_raster_shade(const float* __restrict__ wsB,
               const float* __restrict__ wsW,
               const float* __restrict__ vertices,
               const float* __restrict__ normals,
               const float* __restrict__ colors,
               const int*   __restrict__ faces,
               const float* __restrict__ lpos,
               const float* __restrict__ lint,
               float* __restrict__ out)
{
    __shared__ float sB[NT * 16];       // 16 KB triangle matrices for this batch
    __shared__ int   sBest[8][16];      // per-wave best-triangle per pixel

    const int tid  = threadIdx.x;
    const int b    = blockIdx.x >> 9;          // 512 blocks per batch
    const int blk  = blockIdx.x & 511;
    const int lane = tid & 31;
    const int w    = tid >> 5;

    // warm the lines the shading tail will gather from (global_prefetch_b8)
    __builtin_prefetch(wsW + b * (NT * 4), 0, 1);
    __builtin_prefetch(normals + b * (NV * 3), 0, 1);

    // stage triangle data into LDS
    const float* gB = wsB + b * (NT * 16);
#if USE_ASYNC_LDS
    #pragma unroll
    for (int j = 0; j < 4; ++j) {
        int c = (tid + j * 256) * 4;   // float index of this 16-byte chunk
        __builtin_amdgcn_global_load_async_to_lds_b128(
            (gv4i_p)(gB + c),
            (lv4i_p)(&sB[c]),
            0, 0);
    }
    __builtin_amdgcn_s_wait_asynccnt(0);
#else
    #pragma unroll
    for (int j = 0; j < 16; ++j) sB[tid + j*256] = gB[tid + j*256];
#endif
    __syncthreads();

    const int  pixBase = blk * 128 + w * 16;     // 16 pixels, same image row
    const int  yrow    = pixBase >> 8;
    const int  xcol0   = pixBase & 255;
    const bool hi      = lane >= 16;
    const int  m16     = lane & 15;

    const float px = 2.0f * ((float)(xcol0 + m16) + 0.5f) * (1.0f / 256.0f) - 1.0f;
    const float py = 1.0f - 2.0f * ((float)yrow + 0.5f) * (1.0f / 256.0f);

    // A matrix: lanes 0-15 supply K=0,1 = (px,py); lanes 16-31 supply K=2,3 = (1,0)
    v2f avec;
    avec.x = hi ? 1.0f : px;
    avec.y = hi ? 0.0f : py;

    const int tl       = m16 >> 2;     // triangle slot within group of 4
    const int ci       = m16 & 3;      // column within triangle: b0,b1,b2,z
    const int baseLane = lane & ~3;
    const int kidx     = hi ? 2 : 0;   // B rows K=2,3 in hi half, K=0,1 in lo half

    float zmin[8];
    int   best[8];
    #pragma unroll
    for (int r = 0; r < 8; ++r) { zmin[r] = __builtin_inff(); best[r] = 0x7fffffff; }

    for (int g = 0; g < NT/4; ++g) {
        const int t = g*4 + tl;
        const v2f bvec = *reinterpret_cast<const v2f*>(&sB[t*16 + ci*4 + kidx]);
        v8f cz = {};
        v8f d = __builtin_amdgcn_wmma_f32_16x16x4_f32(
                    false, avec, false, bvec, (short)0, cz, false, false);
        #pragma unroll
        for (int r = 0; r < 8; ++r) {
            float v = d[r];
            unsigned msk = __builtin_amdgcn_ballot_w32(v >= 0.0f);
            bool covered = ((msk >> baseLane) & 7u) == 7u;
            float z  = __shfl(v, baseLane + 3, 32);
            float zc = covered ? z : __builtin_inff();
            if (zc < zmin[r]) { zmin[r] = zc; best[r] = t; }
        }
    }

    // reduce across the 4 triangle slots (lanes differing in bits 2,3),
    // preferring the lowest triangle index on z ties (argmin semantics)
    #pragma unroll
    for (int r = 0; r < 8; ++r) {
        float z  = zmin[r];
        int   bt = best[r];
        #pragma unroll
        for (int off = 4; off <= 8; off <<= 1) {
            float oz = __shfl_xor(z, off, 32);
            int   ot = __shfl_xor(bt, off, 32);
            if (oz < z || (oz == z && ot < bt)) { z = oz; bt = ot; }
        }
        zmin[r] = z; best[r] = bt;
    }

    if ((lane & 15) == 0) {
        #pragma unroll
        for (int r = 0; r < 8; ++r) {
            int m = hi ? (r + 8) : r;
            sBest[w][m] = (zmin[r] < __builtin_inff()) ? best[r] : -1;
        }
    }
    __syncthreads();

    // ---- shading: one pixel per lane (lanes 0-15) ----
    if (lane < 16) {
        const int m     = lane;           // == m16, so px/py above are this pixel's
        const int bestT = sBest[w][m];
        const int pix   = pixBase + m;
        float4 o;
        if (bestT < 0) {
            o = make_float4(0.0f, 0.0f, 0.0f, 0.0f);
        } else {
            const float* Bt = &sB[bestT * 16];
            float bry[3];
            #pragma unroll
            for (int i = 0; i < 3; ++i)
                bry[i] = Bt[i*4 + 0]*px + Bt[i*4 + 1]*py + Bt[i*4 + 2];

            const float* wcp = wsW + (b*NT + bestT)*4;
            float bpc[3]; float den = 0.0f;
            #pragma unroll
            for (int k = 0; k < 3; ++k) {
                float wk = wcp[k];
                float sw = (fabsf(wk) > 1e-8f) ? wk : 1.0f;
                bpc[k] = bry[k] / sw;
                den += bpc[k];
            }
            float sd = (fabsf(den) > 1e-8f) ? den : 1.0f;
            #pragma unroll
            for (int k = 0; k < 3; ++k) bpc[k] /= sd;

            int vi[3] = { faces[bestT*3 + 0], faces[bestT*3 + 1], faces[bestT*3 + 2] };
            float nrm[3] = {0,0,0}, pos[3] = {0,0,0}, col[3] = {0,0,0};
            #pragma unroll
            for (int k = 0; k < 3; ++k) {
                int base = (b*NV + vi[k]) * 3;
                #pragma unroll
                for (int c = 0; c < 3; ++c) {
                    nrm[c] += bpc[k] * normals [base + c];
                    pos[c] += bpc[k] * vertices[base + c];
                    col[c] += bpc[k] * colors  [base + c];
                }
            }
            float nl = sqrtf(nrm[0]*nrm[0] + nrm[1]*nrm[1] + nrm[2]*nrm[2]) + 1e-8f;
            nrm[0] /= nl; nrm[1] /= nl; nrm[2] /= nl;

            float shade[3] = {0,0,0};
            #pragma unroll
            for (int l = 0; l < NL; ++l) {
                float dx = lpos[l*3 + 0] - pos[0];
                float dy = lpos[l*3 + 1] - pos[1];
                float dz = lpos[l*3 + 2] - pos[2];
                float dl = sqrtf(dx*dx + dy*dy + dz*dz) + 1e-8f;
                float nd = (dx*nrm[0] + dy*nrm[1] + dz*nrm[2]) / dl;
                nd = fmaxf(nd, 0.0f);
                #pragma unroll
                for (int c = 0; c < 3; ++c) shade[c] += nd * lint[l*3 + c];
            }
            o = make_float4(col[0]*shade[0], col[1]*shade[1], col[2]*shade[2], 1.0f);
        }
        float4* op = reinterpret_cast<float4*>(out) + (b * (IMG*IMG) + pix);
        *op = o;
    }
}

// ---------------------------------------------------------------------------
// kernel_launch
//   d_in order: vertices, normals, colors, eye, center, world_up,
//               light_positions, light_intensities, faces
//   d_out: (B,H,W,4) float32
//   d_ws layout (floats): [0,4096) clip | [4096,12288) wsB | [12288,14336) wsW
// ---------------------------------------------------------------------------
extern "C" void kernel_launch(void* const* d_in, const int* in_sizes, int n_in,
                              void* d_out, int out_size, void* d_ws, size_t ws_size,
                              hipStream_t stream)
{
    const float* vertices = (const float*)d_in[0];
    const float* normals  = (const float*)d_in[1];
    const float* colors   = (const float*)d_in[2];
    const float* eye      = (const float*)d_in[3];
    const float* center   = (const float*)d_in[4];
    const float* wup      = (const float*)d_in[5];
    const float* lpos     = (const float*)d_in[6];
    const float* lint     = (const float*)d_in[7];
    const int*   faces    = (const int*)  d_in[8];
    float* outp = (float*)d_out;

    float* wsf  = (float*)d_ws;
    float* clip = wsf;                 // NB*NV*4   = 4096 floats
    float* wsB  = wsf + 4096;          // NB*NT*16  = 8192 floats
    float* wsW  = wsf + 12288;         // NB*NT*4   = 2048 floats

    k_transform<<<(NB*NV + 255)/256, 256, 0, stream>>>(vertices, eye, center, wup, clip);
    k_trisetup <<<(NB*NT + 255)/256, 256, 0, stream>>>(clip, faces, wsB, wsW);
    k_raster_shade<<<NB * (IMG*IMG/128), 256, 0, stream>>>(
        wsB, wsW, vertices, normals, colors, faces, lpos, lint, outp);
}